// SkeletonTransfer_46136538694005
// MI455X (gfx1250) — compile-verified
//
#include <hip/hip_runtime.h>

typedef __attribute__((ext_vector_type(2))) float v2f;
typedef __attribute__((ext_vector_type(4))) float v4f;
typedef __attribute__((ext_vector_type(8))) float v8f;

#define B_N 128
#define V_N 50000
#define J_N 64
#define KSTEPS 12500            /* V/4 */
#define KSPLIT 16
#define KCHUNK 782              /* ceil(12500/16) */
#define CT_N 20                 /* 320/16 */
#define NT_N 24                 /* 384/16 */
#define TILES (CT_N*NT_N)       /* 480 */
#define G_N 5                   /* groups of 4 c-tiles */

/* workspace float offsets */
#define OFF_MW   0ull                    /* 16,000,000 */
#define OFF_PART 16000000ull             /* 16*480*256 = 1,966,080 */
#define OFF_WSUM 17966080ull             /* 64 */
#define OFF_MBN  17966144ull             /* 192 */
#define OFF_R    17966336ull             /* 8192*9 = 73,728 */
#define OFF_T    18040064ull             /* 8192*3 = 24,576 */

// ---------------------------------------------------------------------------
// Stage 1: pack combined matrix M (V x 320) into WMMA-A swizzled layout,
// grouped so one wave serves 4 c-tiles with two b128 loads per lane per step.
// M[v, c]:  c<64 -> regressor[c,v];  64<=c<128 -> W[v,c-64];
//           c>=128 -> W[v,j]*bind[v,d]  (j=(c-128)/3, d=(c-128)%3)
// Element index: ((g*KSTEPS + s)*32 + lane)*8 + ci*2 + r
//   holds M[4*s + 2*(lane>=16) + r, (g*4+ci)*16 + (lane&15)].
// ---------------------------------------------------------------------------
__global__ void __launch_bounds__(256)
pack_M(const float* __restrict__ reg, const float* __restrict__ W,
       const float* __restrict__ bind, float* __restrict__ Mw) {
    size_t tid = (size_t)blockIdx.x * blockDim.x + threadIdx.x;
    if (tid >= (size_t)320 * V_N) return;
    unsigned r    = (unsigned)(tid & 1u);
    unsigned ci   = (unsigned)((tid >> 1) & 3u);
    unsigned lane = (unsigned)((tid >> 3) & 31u);
    size_t   gs   = tid >> 8;                 // g*KSTEPS + s
    unsigned g    = (unsigned)(gs / KSTEPS);
    unsigned s    = (unsigned)(gs % KSTEPS);
    unsigned v    = 4u * s + ((lane >> 4) << 1) + r;
    unsigned c    = (g * 4u + ci) * 16u + (lane & 15u);
    float val;
    if (c < 64u) {
        val = reg[(size_t)c * V_N + v];
    } else if (c < 128u) {
        val = W[(size_t)v * J_N + (c - 64u)];
    } else {
        unsigned cc = c - 128u, j = cc / 3u, d = cc - 3u * j;
        val = W[(size_t)v * J_N + j] * bind[(size_t)v * 3u + d];
    }
    Mw[tid] = val;
}

// ---------------------------------------------------------------------------
// Stage 2: per-joint scalars: Wsum[j] = sum_v W[v,j];  mbn[j,d] = sum_v W[v,j]*bind[v,d]
// ---------------------------------------------------------------------------
__global__ void __launch_bounds__(256)
joint_stats(const float* __restrict__ W, const float* __restrict__ bind,
            float* __restrict__ wsum, float* __restrict__ mbn) {
    __shared__ float s0[256], s1[256], s2[256], s3[256];
    int j = blockIdx.x, t = threadIdx.x;
    float a0 = 0.f, a1 = 0.f, a2 = 0.f, a3 = 0.f;
    for (int v = t; v < V_N; v += 256) {
        float w = W[(size_t)v * J_N + j];
        a0 += w;
        a1 += w * bind[(size_t)v * 3 + 0];
        a2 += w * bind[(size_t)v * 3 + 1];
        a3 += w * bind[(size_t)v * 3 + 2];
    }
    s0[t] = a0; s1[t] = a1; s2[t] = a2; s3[t] = a3;
    __syncthreads();
    for (int off = 128; off > 0; off >>= 1) {
        if (t < off) { s0[t] += s0[t+off]; s1[t] += s1[t+off]; s2[t] += s2[t+off]; s3[t] += s3[t+off]; }
        __syncthreads();
    }
    if (t == 0) {
        wsum[j] = s0[0];
        mbn[j*3+0] = s1[0]; mbn[j*3+1] = s2[0]; mbn[j*3+2] = s3[0];
    }
}

// ---------------------------------------------------------------------------
// Stage 3: C[320 x 384] = M^T (320xV) * T (Vx384), n = b*3+d, split-K partials.
// One wave = 64x16 macro-tile (4 c-tiles, shared B fragment); 4 waves/block.
// Per K-step: 2x b128 (A), 2x b32 (B), 4x v_wmma_f32_16x16x4_f32, software
// pipelined with distance-1 prefetch so loads overlap the matrix pipe.
// ---------------------------------------------------------------------------
__global__ void __launch_bounds__(128)
gemm_wmma(const float* __restrict__ Mw, const float* __restrict__ tgt,
          float* __restrict__ part) {
    int wave = threadIdx.x >> 5;
    int lane = threadIdx.x & 31;
    int task = blockIdx.x * 4 + wave;          // 0..119 : g*NT_N + nt
    int z    = blockIdx.y;                     // K-chunk 0..15
    int g    = task / NT_N;
    int nt   = task - g * NT_N;
    int hl   = lane >> 4;                      // lane half
    int col  = lane & 15;

    int n = nt * 16 + col;
    int b = n / 3;
    int d = n - 3 * b;
    const float* tb = tgt + (size_t)b * (V_N * 3) + d;   // stride 3 floats in v

    int s0 = z * KCHUNK;
    int s1 = s0 + KCHUNK; if (s1 > KSTEPS) s1 = KSTEPS;
    int steps = s1 - s0;
    int k0 = 2 * hl;                           // lanes 0-15: k=0,1 ; 16-31: k=2,3

    const v4f* ap = (const v4f*)(Mw + ((size_t)g * KSTEPS + s0) * 256) + lane * 2;

    v8f acc0 = {}, acc1 = {}, acc2 = {}, acc3 = {};

    // prologue loads (steps >= 1 always: KCHUNK chunks never empty)
    v4f f0 = ap[0];
    v4f f1 = ap[1];
    float pb0 = tb[(size_t)(4 * s0 + k0) * 3];
    float pb1 = tb[(size_t)(4 * s0 + k0) * 3 + 3];

    for (int i = 0; i < steps; ++i) {
        v4f  cf0 = f0, cf1 = f1;
        v2f  bv; bv.x = pb0; bv.y = pb1;
        if (i + 1 < steps) {
            const v4f* apn = ap + (size_t)(i + 1) * 64;
            f0  = apn[0];
            f1  = apn[1];
            size_t vv3 = (size_t)(4 * (s0 + i + 1) + k0) * 3;
            pb0 = tb[vv3];
            pb1 = tb[vv3 + 3];
        }
        v2f a0; a0.x = cf0.x; a0.y = cf0.y;
        v2f a1; a1.x = cf0.z; a1.y = cf0.w;
        v2f a2; a2.x = cf1.x; a2.y = cf1.y;
        v2f a3; a3.x = cf1.z; a3.y = cf1.w;
        acc0 = __builtin_amdgcn_wmma_f32_16x16x4_f32(false, a0, false, bv, (short)0, acc0, false, false);
        acc1 = __builtin_amdgcn_wmma_f32_16x16x4_f32(false, a1, false, bv, (short)0, acc1, false, false);
        acc2 = __builtin_amdgcn_wmma_f32_16x16x4_f32(false, a2, false, bv, (short)0, acc2, false, false);
        acc3 = __builtin_amdgcn_wmma_f32_16x16x4_f32(false, a3, false, bv, (short)0, acc3, false, false);
    }

    // store 4 tile partials: tile = (g*4+ci)*NT_N + nt
#pragma unroll
    for (int ci = 0; ci < 4; ++ci) {
        v8f a = (ci == 0) ? acc0 : (ci == 1) ? acc1 : (ci == 2) ? acc2 : acc3;
        int tile = (g * 4 + ci) * NT_N + nt;
        float* op = part + ((size_t)z * TILES + tile) * 256;
#pragma unroll
        for (int r = 0; r < 8; ++r)
            op[(r + hl * 8) * 16 + col] = a[r];
    }
}

// ---------------------------------------------------------------------------
// Stage 4: reduce split-K, form H, Kabsch via Horn quaternion power iteration.
// ---------------------------------------------------------------------------
__device__ __forceinline__ float sumC(const float* part, int c, int n) {
    int ct = c >> 4, row = c & 15, nt = n >> 4, cn = n & 15;
    size_t base = ((size_t)(ct * NT_N + nt)) * 256 + (size_t)row * 16 + cn;
    float s = 0.f;
#pragma unroll
    for (int z = 0; z < KSPLIT; ++z) s += part[(size_t)z * TILES * 256 + base];
    return s;
}

__global__ void __launch_bounds__(256)
kabsch(const float* __restrict__ part, const float* __restrict__ wsum,
       const float* __restrict__ mbn, float* __restrict__ Rg, float* __restrict__ Tg) {
    int idx = blockIdx.x * blockDim.x + threadIdx.x;   // b*64 + j
    if (idx >= B_N * J_N) return;
    int b = idx >> 6, j = idx & 63;

    float jp[3], mt[3], H[3][3];
#pragma unroll
    for (int d = 0; d < 3; ++d) {
        jp[d] = sumC(part, j,        3 * b + d);
        mt[d] = sumC(part, 64 + j,   3 * b + d);
    }
    float inw = 1.0f / wsum[j];
#pragma unroll
    for (int d = 0; d < 3; ++d)
#pragma unroll
        for (int e = 0; e < 3; ++e)
            H[d][e] = sumC(part, 128 + 3 * j + d, 3 * b + e) - mbn[j * 3 + d] * mt[e] * inw;

    float Sxx=H[0][0], Sxy=H[0][1], Sxz=H[0][2];
    float Syx=H[1][0], Syy=H[1][1], Syz=H[1][2];
    float Szx=H[2][0], Szy=H[2][1], Szz=H[2][2];
    float N[4][4] = {
        { Sxx+Syy+Szz, Syz-Szy,      Szx-Sxz,      Sxy-Syx      },
        { Syz-Szy,     Sxx-Syy-Szz,  Sxy+Syx,      Szx+Sxz      },
        { Szx-Sxz,     Sxy+Syx,     -Sxx+Syy-Szz,  Syz+Szy      },
        { Sxy-Syx,     Szx+Sxz,      Syz+Szy,     -Sxx-Syy+Szz  }};
    float bnd = 2.0f*(fabsf(Sxx)+fabsf(Syy)+fabsf(Szz)+fabsf(Sxy)+fabsf(Syx)
                     +fabsf(Sxz)+fabsf(Szx)+fabsf(Syz)+fabsf(Szy)) + 1e-30f;

    float q[4] = {1.f, 0.11f, 0.23f, 0.17f};
    for (int it = 0; it < 60; ++it) {
        float p[4];
#pragma unroll
        for (int r = 0; r < 4; ++r)
            p[r] = N[r][0]*q[0] + N[r][1]*q[1] + N[r][2]*q[2] + N[r][3]*q[3] + bnd*q[r];
        float nn = rsqrtf(p[0]*p[0] + p[1]*p[1] + p[2]*p[2] + p[3]*p[3]);
        q[0]=p[0]*nn; q[1]=p[1]*nn; q[2]=p[2]*nn; q[3]=p[3]*nn;
    }
    float w=q[0], x=q[1], y=q[2], zz=q[3];
    float R[9] = { w*w+x*x-y*y-zz*zz, 2.f*(x*y-w*zz),     2.f*(x*zz+w*y),
                   2.f*(x*y+w*zz),    w*w-x*x+y*y-zz*zz,  2.f*(y*zz-w*x),
                   2.f*(x*zz-w*y),    2.f*(y*zz+w*x),     w*w-x*x-y*y+zz*zz };
#pragma unroll
    for (int i = 0; i < 9; ++i) Rg[(size_t)idx*9 + i] = R[i];
#pragma unroll
    for (int d = 0; d < 3; ++d) Tg[(size_t)idx*3 + d] = jp[d];
}

// ---------------------------------------------------------------------------
// Stage 5: parent-relative transforms, assemble 4x4 output.
// local_R = Rp^T R ; local_t = Rp^T (t - tp) ; root uses global.
// ---------------------------------------------------------------------------
__global__ void __launch_bounds__(256)
compose(const float* __restrict__ Rg, const float* __restrict__ Tg,
        const int* __restrict__ par, float* __restrict__ out) {
    int idx = blockIdx.x * blockDim.x + threadIdx.x;
    if (idx >= B_N * J_N) return;
    int b = idx >> 6, j = idx & 63;
    const float* R = Rg + (size_t)idx * 9;
    const float* t = Tg + (size_t)idx * 3;
    float L[9], lt[3];
    if (j == 0) {
#pragma unroll
        for (int i = 0; i < 9; ++i) L[i] = R[i];
#pragma unroll
        for (int d = 0; d < 3; ++d) lt[d] = t[d];
    } else {
        int p = par[j];
        const float* Rp = Rg + ((size_t)(b * J_N + p)) * 9;
        const float* tp = Tg + ((size_t)(b * J_N + p)) * 3;
        float dt[3] = { t[0]-tp[0], t[1]-tp[1], t[2]-tp[2] };
#pragma unroll
        for (int y = 0; y < 3; ++y) {
#pragma unroll
            for (int z = 0; z < 3; ++z)
                L[y*3+z] = Rp[0*3+y]*R[0*3+z] + Rp[1*3+y]*R[1*3+z] + Rp[2*3+y]*R[2*3+z];
            lt[y] = Rp[0*3+y]*dt[0] + Rp[1*3+y]*dt[1] + Rp[2*3+y]*dt[2];
        }
    }
    float* o = out + (size_t)idx * 16;
#pragma unroll
    for (int r = 0; r < 3; ++r) {
        o[r*4+0] = L[r*3+0]; o[r*4+1] = L[r*3+1]; o[r*4+2] = L[r*3+2]; o[r*4+3] = lt[r];
    }
    o[12] = 0.f; o[13] = 0.f; o[14] = 0.f; o[15] = 1.f;
}

// ---------------------------------------------------------------------------
extern "C" void kernel_launch(void* const* d_in, const int* in_sizes, int n_in,
                              void* d_out, int out_size, void* d_ws, size_t ws_size,
                              hipStream_t stream) {
    const float* tgt  = (const float*)d_in[0];   // (B,V,3)
    const float* reg  = (const float*)d_in[1];   // (J,V)
    const float* W    = (const float*)d_in[2];   // (V,J)
    const float* bind = (const float*)d_in[3];   // (V,3)
    const int*   par  = (const int*)d_in[4];     // (J,)
    float* out = (float*)d_out;
    float* ws  = (float*)d_ws;

    float* Mw   = ws + OFF_MW;
    float* part = ws + OFF_PART;
    float* wsum = ws + OFF_WSUM;
    float* mbn  = ws + OFF_MBN;
    float* Rg   = ws + OFF_R;
    float* Tg   = ws + OFF_T;

    pack_M<<<62500, 256, 0, stream>>>(reg, W, bind, Mw);                 // 320*V/256
    joint_stats<<<J_N, 256, 0, stream>>>(W, bind, wsum, mbn);
    gemm_wmma<<<dim3((G_N * NT_N) / 4, KSPLIT), 128, 0, stream>>>(Mw, tgt, part);
    kabsch<<<(B_N * J_N) / 256, 256, 0, stream>>>(part, wsum, mbn, Rg, Tg);
    compose<<<(B_N * J_N) / 256, 256, 0, stream>>>(Rg, Tg, par, out);
}